// SignalSovereignty_88304527606121
// MI455X (gfx1250) — compile-verified
//
#include <hip/hip_runtime.h>
#include <cstdint>

// SignalSovereignty eval-mode forward for MI455X (gfx1250).
// Bandwidth-bound segment-normalize: roofline ~34us @ 23.3 TB/s.
// CDNA5 features used:
//   * 320KB WGP LDS: whole 132KB group-stat table privatized per block (ds_add_f32)
//   * global_load_async_to_lds_b128/_b32 + s_wait_asynccnt double-buffered streaming
// WMMA intentionally unused: data-dependent scatter-reduce has no matmul structure.

#define KDIM   16
#define GROUPS 1024          // num_groups from the reference setup (python scalar lives on device)
#define EPSV   1e-8f
#define NTPB   256           // threads/block for normalize pass (8 waves)
#define TILE_ROWS 256        // rows per async tile: 16KB x + 1KB gid

// ---------------- CDNA5 async global->LDS helpers ----------------
// Generic pointer to a __shared__ object: low 32 bits == LDS byte offset
// (flat LDS aperture maps by truncation; VDST of async ops wants this offset).
__device__ __forceinline__ unsigned lds_byte_off(const void* p) {
  return (unsigned)(unsigned long long)(uintptr_t)p;
}

__device__ __forceinline__ void async_copy_b128(const float* g, float* l) {
#if defined(__HIP_DEVICE_COMPILE__) && defined(__gfx1250__)
  asm volatile("global_load_async_to_lds_b128 %0, %1, off"
               :: "v"(lds_byte_off(l)), "v"(g) : "memory");
#else
  l[0] = g[0]; l[1] = g[1]; l[2] = g[2]; l[3] = g[3];
#endif
}

__device__ __forceinline__ void async_copy_b32(const int* g, int* l) {
#if defined(__HIP_DEVICE_COMPILE__) && defined(__gfx1250__)
  asm volatile("global_load_async_to_lds_b32 %0, %1, off"
               :: "v"(lds_byte_off(l)), "v"(g) : "memory");
#else
  *l = *g;
#endif
}

__device__ __forceinline__ void wait_async_le5() {
#if defined(__HIP_DEVICE_COMPILE__) && defined(__gfx1250__)
  asm volatile("s_wait_asynccnt 5" ::: "memory");
#endif
}
__device__ __forceinline__ void wait_async_0() {
#if defined(__HIP_DEVICE_COMPILE__) && defined(__gfx1250__)
  asm volatile("s_wait_asynccnt 0" ::: "memory");
#endif
}

// ---------------- kernel 0: zero the accumulator workspace ----------------
__global__ void k_zero(float* __restrict__ p, int n) {
  for (int i = blockIdx.x * blockDim.x + threadIdx.x; i < n; i += gridDim.x * blockDim.x)
    p[i] = 0.0f;
}

// ---------------- kernel 1: LDS-privatized segment reduce ----------------
// ws layout: counts[G] | sums[G*K] | sumsq[G*K]   (132KB, fits CDNA5 WGP LDS)
__global__ __launch_bounds__(512) void k_reduce(const float* __restrict__ x,
                                                const int* __restrict__ gid,
                                                float* __restrict__ gacc, int B) {
  extern __shared__ float lacc[];
  const int ACCN = GROUPS * (1 + 2 * KDIM);
  for (int i = threadIdx.x; i < ACCN; i += blockDim.x) lacc[i] = 0.0f;
  __syncthreads();

  float* lcnt = lacc;
  float* lsum = lacc + GROUPS;
  float* lss  = lacc + GROUPS + GROUPS * KDIM;

  const int nq = B * 4;                       // one float4 (4 of the 16 dims) per quad
  const int stride = gridDim.x * blockDim.x;
  for (int q = blockIdx.x * blockDim.x + threadIdx.x; q < nq; q += stride) {
    const int row = q >> 2;
    const int k4  = (q & 3) * 4;
    const float4 v = *(const float4*)(x + (size_t)row * KDIM + k4);
    const int g = gid[row];
    if ((unsigned)g < (unsigned)GROUPS) {
      float* s  = lsum + g * KDIM + k4;
      float* sq = lss  + g * KDIM + k4;
      atomicAdd(s + 0, v.x); atomicAdd(sq + 0, v.x * v.x);
      atomicAdd(s + 1, v.y); atomicAdd(sq + 1, v.y * v.y);
      atomicAdd(s + 2, v.z); atomicAdd(sq + 2, v.z * v.z);
      atomicAdd(s + 3, v.w); atomicAdd(sq + 3, v.w * v.w);
      if (k4 == 0) atomicAdd(lcnt + g, 1.0f);
    }
  }
  __syncthreads();
  // one dense flush per block: no-return global_atomic_add_f32
  for (int i = threadIdx.x; i < ACCN; i += blockDim.x) atomicAdd(&gacc[i], lacc[i]);
}

// ---------------- kernel 2: finalize mean / inv(std+eps) tables in place ----
// counts>1 -> group stats; else running-stat fallback (matches reference).
__global__ void k_finalize(float* __restrict__ ws,
                           const float* __restrict__ rmean,
                           const float* __restrict__ rvar) {
  const int idx = blockIdx.x * blockDim.x + threadIdx.x;
  if (idx >= GROUPS * KDIM) return;
  const int g = idx >> 4;
  const int k = idx & 15;
  float* cnt  = ws;
  float* sums = ws + GROUPS;
  float* ssq  = ws + GROUPS + GROUPS * KDIM;

  const float c  = cnt[g];
  const float cc = fmaxf(c, 1.0f);
  const float s  = sums[idx];
  const float sq = ssq[idx];
  float mean = s / cc;
  float var  = fmaxf(sq / cc - mean * mean, 0.0f);
  float inv  = 1.0f / (sqrtf(var) + EPSV);
  if (c <= 1.0f) {                    // singleton (or empty) group fallback
    mean = rmean[k];
    inv  = 1.0f / (sqrtf(rvar[k]) + EPSV);
  }
  sums[idx] = mean;                   // reuse sums buffer as mean table
  ssq[idx]  = inv;                    // reuse sumsq buffer as inv table
}

// ---------------- kernel 3: async-double-buffered normalize ----------------
__device__ __forceinline__ void stage_tile(const float* __restrict__ x,
                                           const int* __restrict__ gid,
                                           int tile, float* lx, int* lg) {
  const float* gx = x + (size_t)tile * (TILE_ROWS * KDIM);
#pragma unroll
  for (int i = 0; i < 4; ++i) {                       // 4 x b128 per lane
    const int fi = (threadIdx.x + i * NTPB) * 4;
    async_copy_b128(gx + fi, lx + fi);
  }
  async_copy_b32(gid + (size_t)tile * TILE_ROWS + threadIdx.x, lg + threadIdx.x);
}                                                      // = 5 async ops per wave

__global__ __launch_bounds__(NTPB) void k_normalize(const float* __restrict__ x,
                                                    const int* __restrict__ gid,
                                                    const float* __restrict__ ws,
                                                    float* __restrict__ out, int B) {
  __shared__ alignas(16) float xbuf[2][TILE_ROWS * KDIM];  // 2 x 16KB
  __shared__ int gbuf[2][TILE_ROWS];                       // 2 x 1KB
  const float* meant = ws + GROUPS;
  const float* invt  = ws + GROUPS + GROUPS * KDIM;

  const int ntiles = B / TILE_ROWS;
  int t = (int)blockIdx.x;
  int slot = 0;
  if (t < ntiles) stage_tile(x, gid, t, xbuf[0], gbuf[0]);
  while (t < ntiles) {
    const int nt = t + (int)gridDim.x;
    const bool more = (nt < ntiles);
    if (more) stage_tile(x, gid, nt, xbuf[slot ^ 1], gbuf[slot ^ 1]);
    if (more) wait_async_le5(); else wait_async_0();   // tile t complete (per wave)
    __syncthreads();                                   // all waves' parts landed
#pragma unroll
    for (int i = 0; i < 4; ++i) {
      const int q  = threadIdx.x + i * NTPB;           // [0, 1024) quads in tile
      const int rl = q >> 2;
      const int k4 = (q & 3) * 4;
      const int g  = gbuf[slot][rl];
      const float4 v  = *(const float4*)&xbuf[slot][q * 4];
      const float4 m  = *(const float4*)(meant + (size_t)g * KDIM + k4);
      const float4 iv = *(const float4*)(invt  + (size_t)g * KDIM + k4);
      float4 o;
      o.x = (v.x - m.x) * iv.x;
      o.y = (v.y - m.y) * iv.y;
      o.z = (v.z - m.z) * iv.z;
      o.w = (v.w - m.w) * iv.w;
      *(float4*)(out + (size_t)t * (TILE_ROWS * KDIM) + q * 4) = o;
    }
    __syncthreads();                                   // slot consumed: safe to reuse
    t = nt;
    slot ^= 1;
  }

  // remainder rows (B % TILE_ROWS; zero for B=4M) — plain path
  const int remQ0 = ntiles * TILE_ROWS * 4;
  for (int q = remQ0 + (int)(blockIdx.x * blockDim.x + threadIdx.x); q < B * 4;
       q += (int)(gridDim.x * blockDim.x)) {
    const int row = q >> 2;
    const int k4  = (q & 3) * 4;
    const float4 v = *(const float4*)(x + (size_t)row * KDIM + k4);
    const int g = gid[row];
    const float4 m  = *(const float4*)(meant + (size_t)g * KDIM + k4);
    const float4 iv = *(const float4*)(invt  + (size_t)g * KDIM + k4);
    float4 o;
    o.x = (v.x - m.x) * iv.x;
    o.y = (v.y - m.y) * iv.y;
    o.z = (v.z - m.z) * iv.z;
    o.w = (v.w - m.w) * iv.w;
    *(float4*)(out + (size_t)row * KDIM + k4) = o;
  }
}

// ---------------- host launcher ----------------
extern "C" void kernel_launch(void* const* d_in, const int* in_sizes, int n_in,
                              void* d_out, int out_size, void* d_ws, size_t ws_size,
                              hipStream_t stream) {
  (void)n_in; (void)out_size; (void)ws_size;
  const float* x     = (const float*)d_in[0];   // [B,16]
  // d_in[1] = weights: unused by the reference output
  const float* rmean = (const float*)d_in[2];   // [16]
  const float* rvar  = (const float*)d_in[3];   // [16]
  const int*   gid   = (const int*)d_in[4];     // [B]
  float* out = (float*)d_out;
  float* ws  = (float*)d_ws;                    // counts[G] | sums[G*K] | sumsq[G*K]

  const int B    = in_sizes[0] / KDIM;
  const int ACCN = GROUPS * (1 + 2 * KDIM);     // 33792 floats = 132KB

  // Opt in to >64KB dynamic LDS (CDNA5 WGP has 320KB). Idempotent; not a
  // stream op, so it does not break graph capture. Errors ignored.
  (void)hipFuncSetAttribute((const void*)k_reduce,
                            hipFuncAttributeMaxDynamicSharedMemorySize,
                            ACCN * (int)sizeof(float));

  k_zero<<<64, 256, 0, stream>>>(ws, ACCN);
  k_reduce<<<256, 512, ACCN * sizeof(float), stream>>>(x, gid, ws, B);
  k_finalize<<<(GROUPS * KDIM + 255) / 256, 256, 0, stream>>>(ws, rmean, rvar);
  k_normalize<<<1024, NTPB, 0, stream>>>(x, gid, ws, out, B);
}